// GNNLocalizationModel_8942121910958
// MI455X (gfx1250) — compile-verified
//
#include <hip/hip_runtime.h>
#include <hip/hip_bf16.h>
#include <math.h>

typedef __attribute__((ext_vector_type(2))) float v2f;
typedef __attribute__((ext_vector_type(8))) float v8f;

#define HC 256
#define NHEAD 4
#define NEG_SLOPE 0.2f
#define NGRAPH 64

// ---------------------------------------------------------------------------
// Utility fills
// ---------------------------------------------------------------------------
__global__ void fill_f32(float* __restrict__ p, float v, int n) {
  int i = blockIdx.x * blockDim.x + threadIdx.x;
  if (i < n) p[i] = v;
}

// ---------------------------------------------------------------------------
// C[M,Nn] = A[M,K] @ B[K,Nn] via V_WMMA_F32_16X16X4_F32.
// One wave computes a 16x64 tile (A fragment reused across 4 WMMAs / k-step).
// A-frag (16x4 f32): lane l holds A[m=l%16][k+2*(l/16) + {0,1}]
// B-frag (4x16 f32): lane l holds B[k+2*(l/16) + {0,1}][n=l%16]
// D (16x16 f32):     vgpr r, lane l -> D[r + 8*(l/16)][l%16]
// ---------------------------------------------------------------------------
__global__ void gemm_wmma_f32(const float* __restrict__ A,
                              const float* __restrict__ B,
                              float* __restrict__ C,
                              int M, int K, int Nn) {
  int gwave = (int)((blockIdx.x * blockDim.x + threadIdx.x) >> 5);
  int lane  = threadIdx.x & 31;
  int nGroups = Nn >> 6;                 // 64-wide column groups
  int mTile  = gwave / nGroups;
  int nGroup = gwave - mTile * nGroups;
  if (mTile * 16 >= M) return;           // wave-uniform: EXEC stays all-ones

  const int row = mTile * 16 + (lane & 15);
  const int kh  = (lane >> 4) << 1;      // 0 or 2
  const int col = (nGroup << 6) + (lane & 15);

  v8f acc0 = {}, acc1 = {}, acc2 = {}, acc3 = {};
  const float* Arow = A + (size_t)row * K;
  for (int k = 0; k < K; k += 4) {
    v2f a;
    a.x = Arow[k + kh];
    a.y = Arow[k + kh + 1];
    const float* B0 = B + (size_t)(k + kh) * Nn + col;
    const float* B1 = B0 + Nn;
    v2f b0, b1, b2, b3;
    b0.x = B0[0];  b0.y = B1[0];
    b1.x = B0[16]; b1.y = B1[16];
    b2.x = B0[32]; b2.y = B1[32];
    b3.x = B0[48]; b3.y = B1[48];
    acc0 = __builtin_amdgcn_wmma_f32_16x16x4_f32(false, a, false, b0, (short)0, acc0, false, false);
    acc1 = __builtin_amdgcn_wmma_f32_16x16x4_f32(false, a, false, b1, (short)0, acc1, false, false);
    acc2 = __builtin_amdgcn_wmma_f32_16x16x4_f32(false, a, false, b2, (short)0, acc2, false, false);
    acc3 = __builtin_amdgcn_wmma_f32_16x16x4_f32(false, a, false, b3, (short)0, acc3, false, false);
  }
  int mBase = mTile * 16 + ((lane >> 4) << 3);
  int cBase = (nGroup << 6) + (lane & 15);
  for (int r = 0; r < 8; ++r) {
    float* Crow = C + (size_t)(mBase + r) * Nn + cBase;
    Crow[0]  = acc0[r];
    Crow[16] = acc1[r];
    Crow[32] = acc2[r];
    Crow[48] = acc3[r];
  }
}

// ---------------------------------------------------------------------------
// Edge attention logits + segment max.
// Wave per edge; lane owns 8 contiguous channels (head = lane>>3).
// ---------------------------------------------------------------------------
__device__ __forceinline__ float lrelu(float v) {
  return v > 0.0f ? v : NEG_SLOPE * v;
}

__device__ __forceinline__ void atomicMaxF32(float* addr, float val) {
  if (val >= 0.0f)
    atomicMax((int*)addr, __float_as_int(val));
  else
    atomicMin((unsigned int*)addr, __float_as_uint(val));
}

__global__ void edge_attention(const float* __restrict__ xl,
                               const float* __restrict__ xr,
                               const int* __restrict__ src_e,
                               const int* __restrict__ dst_e,
                               int E_, int Etot,
                               const float* __restrict__ att,
                               float* __restrict__ e_out,
                               float* __restrict__ emax) {
  int wave = (int)((blockIdx.x * blockDim.x + threadIdx.x) >> 5);
  int lane = threadIdx.x & 31;
  if (wave >= Etot) return;
  int s = (wave < E_) ? src_e[wave] : (wave - E_);   // implicit self-loop
  int d = (wave < E_) ? dst_e[wave] : (wave - E_);

  const float4* pl = (const float4*)(xl + (size_t)s * HC) + lane * 2;
  const float4* pr = (const float4*)(xr + (size_t)d * HC) + lane * 2;
  const float4* pa = (const float4*)att + lane * 2;
  float4 l0 = pl[0], l1 = pl[1];
  float4 r0 = pr[0], r1 = pr[1];
  float4 a0 = pa[0], a1 = pa[1];

  float p = lrelu(l0.x + r0.x) * a0.x + lrelu(l0.y + r0.y) * a0.y +
            lrelu(l0.z + r0.z) * a0.z + lrelu(l0.w + r0.w) * a0.w +
            lrelu(l1.x + r1.x) * a1.x + lrelu(l1.y + r1.y) * a1.y +
            lrelu(l1.z + r1.z) * a1.z + lrelu(l1.w + r1.w) * a1.w;
  // reduce across the 8 lanes of each head group
  p += __shfl_xor(p, 1, 32);
  p += __shfl_xor(p, 2, 32);
  p += __shfl_xor(p, 4, 32);

  int h = lane >> 3;
  if ((lane & 7) == 0) {
    e_out[wave * NHEAD + h] = p;
    atomicMaxF32(&emax[d * NHEAD + h], p);
  }
}

// ---------------------------------------------------------------------------
// ex = exp(e - emax[dst]); denom[dst] += ex.   Thread per (edge, head).
// ---------------------------------------------------------------------------
__global__ void softmax_num(const int* __restrict__ dst_e, int E_, int Etot,
                            float* __restrict__ e,
                            const float* __restrict__ emax,
                            float* __restrict__ denom) {
  int idx = blockIdx.x * blockDim.x + threadIdx.x;
  if (idx >= Etot * NHEAD) return;
  int ed = idx >> 2, h = idx & 3;
  int d = (ed < E_) ? dst_e[ed] : (ed - E_);
  float v = __expf(e[idx] - emax[d * NHEAD + h]);
  e[idx] = v;
  atomicAdd(&denom[d * NHEAD + h], v);
}

// ---------------------------------------------------------------------------
// out[dst] += alpha * xl[src].   Wave per edge, 8 channels / lane.
// ---------------------------------------------------------------------------
__global__ void aggregate(const float* __restrict__ xl,
                          const int* __restrict__ src_e,
                          const int* __restrict__ dst_e,
                          int E_, int Etot,
                          const float* __restrict__ e,
                          const float* __restrict__ denom,
                          float* __restrict__ out) {
  int wave = (int)((blockIdx.x * blockDim.x + threadIdx.x) >> 5);
  int lane = threadIdx.x & 31;
  if (wave >= Etot) return;
  int s = (wave < E_) ? src_e[wave] : (wave - E_);
  int d = (wave < E_) ? dst_e[wave] : (wave - E_);
  int h = lane >> 3;
  float alpha = e[wave * NHEAD + h] / denom[d * NHEAD + h];

  const float4* pl = (const float4*)(xl + (size_t)s * HC) + lane * 2;
  float4 l0 = pl[0], l1 = pl[1];
  float* po = out + (size_t)d * HC + lane * 8;
  atomicAdd(po + 0, alpha * l0.x);
  atomicAdd(po + 1, alpha * l0.y);
  atomicAdd(po + 2, alpha * l0.z);
  atomicAdd(po + 3, alpha * l0.w);
  atomicAdd(po + 4, alpha * l1.x);
  atomicAdd(po + 5, alpha * l1.y);
  atomicAdd(po + 6, alpha * l1.z);
  atomicAdd(po + 7, alpha * l1.w);
}

__global__ void bias_relu(float* __restrict__ p, const float* __restrict__ bias, int total) {
  int i = blockIdx.x * blockDim.x + threadIdx.x;
  if (i < total) {
    float v = p[i] + bias[i & (HC - 1)];
    p[i] = v > 0.0f ? v : 0.0f;
  }
}

// ---------------------------------------------------------------------------
// Global mean pool: block per node, thread per channel.
// ---------------------------------------------------------------------------
__global__ void pool_sum(const float* __restrict__ h, const int* __restrict__ batch,
                         float* __restrict__ gsum, float* __restrict__ gcnt, int Nn) {
  int n = blockIdx.x;
  if (n >= Nn) return;
  int b = batch[n];
  int c = threadIdx.x;
  atomicAdd(&gsum[b * HC + c], h[(size_t)n * HC + c]);
  if (c == 0) atomicAdd(&gcnt[b], 1.0f);
}

// ---------------------------------------------------------------------------
// MLP head: hg = gsum/cnt; z1=relu(hg@mW1+mb1); z2=relu(z1@mW2+mb2); out=z2@mW3+mb3.
// Single block of 256 threads; tiny FLOPs.
// ---------------------------------------------------------------------------
__global__ void mlp_head(const float* __restrict__ gsum, const float* __restrict__ gcnt,
                         const float* __restrict__ mW1, const float* __restrict__ mb1,
                         const float* __restrict__ mW2, const float* __restrict__ mb2,
                         const float* __restrict__ mW3, const float* __restrict__ mb3,
                         float* __restrict__ out) {
  __shared__ float z1[NGRAPH * 64];
  __shared__ float z2[NGRAPH * 32];
  int t = threadIdx.x;
  for (int i = t; i < NGRAPH * 64; i += 256) {
    int g = i >> 6, j = i & 63;
    float rinv = 1.0f / fmaxf(gcnt[g], 1.0f);
    float acc = mb1[j];
    for (int c = 0; c < HC; ++c)
      acc += gsum[g * HC + c] * rinv * mW1[c * 64 + j];
    z1[i] = fmaxf(acc, 0.0f);
  }
  __syncthreads();
  for (int i = t; i < NGRAPH * 32; i += 256) {
    int g = i >> 5, j = i & 31;
    float acc = mb2[j];
    for (int c = 0; c < 64; ++c)
      acc += z1[g * 64 + c] * mW2[c * 32 + j];
    z2[i] = fmaxf(acc, 0.0f);
  }
  __syncthreads();
  for (int i = t; i < NGRAPH * 2; i += 256) {
    int g = i >> 1, j = i & 1;
    float acc = mb3[j];
    for (int c = 0; c < 32; ++c)
      acc += z2[g * 32 + c] * mW3[c * 2 + j];
    out[i] = acc;
  }
}

// ---------------------------------------------------------------------------
// Host-side orchestration
// ---------------------------------------------------------------------------
extern "C" void kernel_launch(void* const* d_in, const int* in_sizes, int n_in,
                              void* d_out, int out_size, void* d_ws, size_t ws_size,
                              hipStream_t stream) {
  const float* x    = (const float*)d_in[0];
  const int*   ei   = (const int*)d_in[1];     // [2, E] flattened
  const int*   bat  = (const int*)d_in[2];
  const float* W1l  = (const float*)d_in[3];
  const float* W1r  = (const float*)d_in[4];
  const float* att1 = (const float*)d_in[5];
  const float* b1   = (const float*)d_in[6];
  const float* W2l  = (const float*)d_in[7];
  const float* W2r  = (const float*)d_in[8];
  const float* att2 = (const float*)d_in[9];
  const float* b2   = (const float*)d_in[10];
  const float* mW1  = (const float*)d_in[11];
  const float* mb1  = (const float*)d_in[12];
  const float* mW2  = (const float*)d_in[13];
  const float* mb2  = (const float*)d_in[14];
  const float* mW3  = (const float*)d_in[15];
  const float* mb3  = (const float*)d_in[16];

  const int N    = in_sizes[0] / 16;   // 50000
  const int E    = in_sizes[1] / 2;    // 800000
  const int Etot = E + N;              // self-loops appended implicitly
  const int* src_e = ei;
  const int* dst_e = ei + E;

  float* ws   = (float*)d_ws;
  float* xl   = ws;
  float* xr   = xl + (size_t)N * HC;
  float* hb   = xr + (size_t)N * HC;   // accumulator / hidden features
  float* ebuf = hb + (size_t)N * HC;   // [Etot, NHEAD]
  float* emax = ebuf + (size_t)Etot * NHEAD;
  float* denom = emax + (size_t)N * NHEAD;
  float* gsum = denom + (size_t)N * NHEAD;
  float* gcnt = gsum + NGRAPH * HC;

  const int TB = 256;
  const int nNodeElems = N * HC;
  dim3 blkFillNode((nNodeElems + TB - 1) / TB);
  dim3 blkFillNH((N * NHEAD + TB - 1) / TB);
  // GEMM grid: (M/16) mTiles * (256/64) nGroups, 8 waves per block
  int tiles = ((N + 15) / 16) * (HC / 64);
  dim3 blkGemm((tiles + 7) / 8);
  dim3 blkEdge((Etot + 7) / 8);                 // wave per edge, 8 waves/block
  dim3 blkSoft((Etot * NHEAD + TB - 1) / TB);

  // ----- Layer 1 -----
  gemm_wmma_f32<<<blkGemm, TB, 0, stream>>>(x, W1l, xl, N, 16, HC);
  gemm_wmma_f32<<<blkGemm, TB, 0, stream>>>(x, W1r, xr, N, 16, HC);
  fill_f32<<<blkFillNode, TB, 0, stream>>>(hb, 0.0f, nNodeElems);
  fill_f32<<<blkFillNH, TB, 0, stream>>>(emax, -INFINITY, N * NHEAD);
  fill_f32<<<blkFillNH, TB, 0, stream>>>(denom, 0.0f, N * NHEAD);
  edge_attention<<<blkEdge, TB, 0, stream>>>(xl, xr, src_e, dst_e, E, Etot, att1, ebuf, emax);
  softmax_num<<<blkSoft, TB, 0, stream>>>(dst_e, E, Etot, ebuf, emax, denom);
  aggregate<<<blkEdge, TB, 0, stream>>>(xl, src_e, dst_e, E, Etot, ebuf, denom, hb);
  bias_relu<<<blkFillNode, TB, 0, stream>>>(hb, b1, nNodeElems);

  // ----- Layer 2 ----- (GEMMs read hb, then hb is recycled as accumulator)
  gemm_wmma_f32<<<blkGemm, TB, 0, stream>>>(hb, W2l, xl, N, HC, HC);
  gemm_wmma_f32<<<blkGemm, TB, 0, stream>>>(hb, W2r, xr, N, HC, HC);
  fill_f32<<<blkFillNode, TB, 0, stream>>>(hb, 0.0f, nNodeElems);
  fill_f32<<<blkFillNH, TB, 0, stream>>>(emax, -INFINITY, N * NHEAD);
  fill_f32<<<blkFillNH, TB, 0, stream>>>(denom, 0.0f, N * NHEAD);
  edge_attention<<<blkEdge, TB, 0, stream>>>(xl, xr, src_e, dst_e, E, Etot, att2, ebuf, emax);
  softmax_num<<<blkSoft, TB, 0, stream>>>(dst_e, E, Etot, ebuf, emax, denom);
  aggregate<<<blkEdge, TB, 0, stream>>>(xl, src_e, dst_e, E, Etot, ebuf, denom, hb);
  bias_relu<<<blkFillNode, TB, 0, stream>>>(hb, b2, nNodeElems);

  // ----- Pool + MLP head -----
  fill_f32<<<dim3((NGRAPH * HC + TB - 1) / TB), TB, 0, stream>>>(gsum, 0.0f, NGRAPH * HC);
  fill_f32<<<dim3(1), TB, 0, stream>>>(gcnt, 0.0f, NGRAPH);
  pool_sum<<<dim3(N), HC, 0, stream>>>(hb, bat, gsum, gcnt, N);
  mlp_head<<<dim3(1), TB, 0, stream>>>(gsum, gcnt, mW1, mb1, mW2, mb2, mW3, mb3,
                                       (float*)d_out);
}